// FeatureVolume_43619687858635
// MI455X (gfx1250) — compile-verified
//
#include <hip/hip_runtime.h>
#include <cmath>

typedef float v2f __attribute__((ext_vector_type(2)));
typedef float v8f __attribute__((ext_vector_type(8)));

#define RESO 48
#define PPTS (RESO * RESO * RESO)   // 110592
#define BATCH 2
#define NV 4
#define CH 32
#define IMH 128
#define IMW 160
#define WAVES_PER_BLOCK 8
#define TILES_PER_B (PPTS / 16)     // 6912

__global__ __launch_bounds__(256) void FeatureVolume_43619687858635_kernel(
    const float* __restrict__ feats, const float* __restrict__ poses,
    const float* __restrict__ w1, const float* __restrict__ b1,
    const float* __restrict__ w2, const float* __restrict__ b2,
    const float* __restrict__ w3, const float* __restrict__ b3,
    float* __restrict__ out)
{
    __shared__ float s_w1[32 * 32];      // [k][n]
    __shared__ float s_w2[32 * 16];      // [k][n]
    __shared__ float s_w3[16 * 16];      // [k][n], cols 8..15 zero-padded
    __shared__ float s_b1[32];
    __shared__ float s_b2[16];
    __shared__ float s_b3[16];
    __shared__ float s_h[WAVES_PER_BLOCK][16 * 32];  // per-wave staging [point][ch]
    __shared__ float s_mask[WAVES_PER_BLOCK][16];
    __shared__ float s_S[WAVES_PER_BLOCK][16];

    const int tid = threadIdx.x;
    for (int i = tid; i < 1024; i += 256) s_w1[i] = w1[i];
    for (int i = tid; i < 512; i += 256)  s_w2[i] = w2[i];
    if (tid < 256) {
        int k = tid >> 4, n = tid & 15;
        s_w3[tid] = (n < 8) ? w3[k * 8 + n] : 0.0f;
    }
    if (tid < 32) s_b1[tid] = b1[tid];
    if (tid < 16) s_b2[tid] = b2[tid];
    if (tid < 16) s_b3[tid] = (tid < 8) ? b3[tid] : 0.0f;
    __syncthreads();

    const int wave = tid >> 5;
    const int lane = tid & 31;
    const int m    = lane & 15;        // point-in-tile (lo/hi halves duplicate)
    const bool hi  = lane >= 16;
    const int koff0 = hi ? 2 : 0;
    const int koff1 = hi ? 3 : 1;

    const int tile = blockIdx.x * WAVES_PER_BLOCK + wave;   // < B*TILES_PER_B
    const int b    = tile / TILES_PER_B;
    const int p0   = (tile % TILES_PER_B) * 16;

    // grid point coordinates (meshgrid 'ij': x slowest, z fastest)
    const int pg  = p0 + m;
    const int xi  = pg / (RESO * RESO);
    const int rem = pg % (RESO * RESO);
    const int yi  = rem / RESO;
    const int zi  = rem % RESO;
    const float sc = 2.0f / (float)(RESO - 1);
    const float X = xi * sc - 1.0f;
    const float Y = yi * sc - 1.0f;
    const float Z = zi * sc - 1.0f;

    float T1[8], T2[8];
    #pragma unroll
    for (int r = 0; r < 8; ++r) { T1[r] = 0.0f; T2[r] = 0.0f; }
    float Ssum = 0.0f;

    float* hl = &s_h[wave][0];

    for (int v = 0; v < NV; ++v) {
        const float* pm = poses + (size_t)(b * NV + v) * 16;
        const float r0 = pm[0] * X + pm[1] * Y + pm[2]  * Z + pm[3];
        const float r1 = pm[4] * X + pm[5] * Y + pm[6]  * Z + pm[7];
        const float r2 = pm[8] * X + pm[9] * Y + pm[10] * Z + pm[11];
        const float mask_d = (r2 > 0.0f) ? 1.0f : 0.0f;
        const float gx = r0 / r2;
        const float gy = r1 / r2;
        const float ix = (gx + 1.0f) * 0.5f * (float)(IMW - 1);
        const float iy = (gy + 1.0f) * 0.5f * (float)(IMH - 1);
        const float ix0 = floorf(ix), iy0 = floorf(iy);
        const float wx1 = ix - ix0, wy1 = iy - iy0;
        const float wx0 = 1.0f - wx1, wy0 = 1.0f - wy1;
        const float w_nw = wx0 * wy0, w_ne = wx1 * wy0;
        const float w_sw = wx0 * wy1, w_se = wx1 * wy1;
        const int cx0 = (int)fminf(fmaxf(ix0,        0.0f), (float)(IMW - 1));
        const int cx1 = (int)fminf(fmaxf(ix0 + 1.0f, 0.0f), (float)(IMW - 1));
        const int cy0 = (int)fminf(fmaxf(iy0,        0.0f), (float)(IMH - 1));
        const int cy1 = (int)fminf(fmaxf(iy0 + 1.0f, 0.0f), (float)(IMH - 1));
        const float inb = (ix >= 0.0f && ix <= (float)(IMW - 1) &&
                           iy >= 0.0f && iy <= (float)(IMH - 1)) ? 1.0f : 0.0f;
        const float mk = inb * mask_d;

        const float* img = feats + (size_t)(b * NV + v) * CH * IMH * IMW;
        __builtin_prefetch(img, 0, 1);
        const int o_nw = cy0 * IMW + cx0, o_ne = cy0 * IMW + cx1;
        const int o_sw = cy1 * IMW + cx0, o_se = cy1 * IMW + cx1;

        // ---- layer 1: vf(16x32) @ w1(32x32) via 8 K-chunks x 2 N-tiles ----
        v8f acc0 = {0.f,0.f,0.f,0.f,0.f,0.f,0.f,0.f};
        v8f acc1 = {0.f,0.f,0.f,0.f,0.f,0.f,0.f,0.f};
        #pragma unroll
        for (int kc = 0; kc < 8; ++kc) {
            const int c0 = 4 * kc + koff0;
            const int c1 = 4 * kc + koff1;
            const float* ch0 = img + (size_t)c0 * IMH * IMW;
            const float* ch1 = img + (size_t)c1 * IMH * IMW;
            v2f a;
            a.x = w_nw * ch0[o_nw] + w_ne * ch0[o_ne] + w_sw * ch0[o_sw] + w_se * ch0[o_se];
            a.y = w_nw * ch1[o_nw] + w_ne * ch1[o_ne] + w_sw * ch1[o_sw] + w_se * ch1[o_se];
            v2f bf0, bf1;
            bf0.x = s_w1[c0 * 32 + m];
            bf0.y = s_w1[c1 * 32 + m];
            bf1.x = s_w1[c0 * 32 + 16 + m];
            bf1.y = s_w1[c1 * 32 + 16 + m];
            acc0 = __builtin_amdgcn_wmma_f32_16x16x4_f32(false, a, false, bf0,
                                                         (short)0, acc0, false, false);
            acc1 = __builtin_amdgcn_wmma_f32_16x16x4_f32(false, a, false, bf1,
                                                         (short)0, acc1, false, false);
        }
        // bias + relu -> stage h1[16][32] in LDS (intra-wave, DS ops are in-order)
        #pragma unroll
        for (int r = 0; r < 8; ++r) {
            const int pt = r + (hi ? 8 : 0);
            hl[pt * 32 + m]      = fmaxf(acc0[r] + s_b1[m],      0.0f);
            hl[pt * 32 + 16 + m] = fmaxf(acc1[r] + s_b1[16 + m], 0.0f);
        }
        __builtin_amdgcn_wave_barrier();

        // ---- layer 2: h1(16x32) @ w2(32x16) ----
        v8f acc2 = {0.f,0.f,0.f,0.f,0.f,0.f,0.f,0.f};
        #pragma unroll
        for (int kc = 0; kc < 8; ++kc) {
            const int c0 = 4 * kc + koff0;
            const int c1 = 4 * kc + koff1;
            v2f a2; a2.x = hl[m * 32 + c0]; a2.y = hl[m * 32 + c1];
            v2f bf; bf.x = s_w2[c0 * 16 + m]; bf.y = s_w2[c1 * 16 + m];
            acc2 = __builtin_amdgcn_wmma_f32_16x16x4_f32(false, a2, false, bf,
                                                         (short)0, acc2, false, false);
        }
        __builtin_amdgcn_wave_barrier();
        #pragma unroll
        for (int r = 0; r < 8; ++r) {
            const int pt = r + (hi ? 8 : 0);
            hl[pt * 32 + m] = fmaxf(acc2[r] + s_b2[m], 0.0f);   // h2 in cols 0..15
        }
        __builtin_amdgcn_wave_barrier();

        // ---- layer 3: h2(16x16) @ w3(16x16, zero-padded) ----
        v8f acc3 = {0.f,0.f,0.f,0.f,0.f,0.f,0.f,0.f};
        #pragma unroll
        for (int kc = 0; kc < 4; ++kc) {
            const int c0 = 4 * kc + koff0;
            const int c1 = 4 * kc + koff1;
            v2f a3; a3.x = hl[m * 32 + c0]; a3.y = hl[m * 32 + c1];
            v2f bf; bf.x = s_w3[c0 * 16 + m]; bf.y = s_w3[c1 * 16 + m];
            acc3 = __builtin_amdgcn_wmma_f32_16x16x4_f32(false, a3, false, bf,
                                                         (short)0, acc3, false, false);
        }

        // ---- masked accumulation over views ----
        s_mask[wave][m] = mk;                // lo/hi write same value
        __builtin_amdgcn_wave_barrier();
        #pragma unroll
        for (int r = 0; r < 8; ++r) {
            const int pt = r + (hi ? 8 : 0);
            const float mm = s_mask[wave][pt];
            const float c  = acc3[r] + s_b3[m];
            T1[r] += mm * c;
            T2[r] += mm * c * c;
        }
        Ssum += mk;
        __builtin_amdgcn_wave_barrier();
    }

    // ---- mean / var and scatter to (B,16,Z,Y,X) ----
    s_S[wave][m] = Ssum;
    __builtin_amdgcn_wave_barrier();
    if (m < 8) {
        #pragma unroll
        for (int r = 0; r < 8; ++r) {
            const int pt = r + (hi ? 8 : 0);
            const float S   = s_S[wave][pt];
            const float inv = 1.0f / (S + 1e-8f);
            const float mean = T1[r] * inv;
            const float A2   = T2[r] * inv;
            const float Ws   = S * inv;
            const float var  = A2 - mean * mean * (2.0f - Ws);
            const int pgp  = p0 + pt;
            const int xo   = pgp / (RESO * RESO);
            const int rm   = pgp % (RESO * RESO);
            const int yo   = rm / RESO;
            const int zo   = rm % RESO;
            const size_t base = ((((size_t)b * 16 + m) * RESO + zo) * RESO + yo) * RESO + xo;
            out[base]                          = mean;
            out[base + (size_t)8 * PPTS]       = var;
        }
    }
}

extern "C" void kernel_launch(void* const* d_in, const int* in_sizes, int n_in,
                              void* d_out, int out_size, void* d_ws, size_t ws_size,
                              hipStream_t stream) {
    const float* feats = (const float*)d_in[0];
    const float* poses = (const float*)d_in[1];
    const float* w1    = (const float*)d_in[2];
    const float* b1    = (const float*)d_in[3];
    const float* w2    = (const float*)d_in[4];
    const float* b2    = (const float*)d_in[5];
    const float* w3    = (const float*)d_in[6];
    const float* b3    = (const float*)d_in[7];
    float* out = (float*)d_out;

    const int total_tiles = BATCH * TILES_PER_B;            // 13824
    const int blocks = total_tiles / WAVES_PER_BLOCK;       // 1728
    FeatureVolume_43619687858635_kernel<<<blocks, 256, 0, stream>>>(
        feats, poses, w1, b1, w2, b2, w3, b3, out);
}